// MSDeformableAttention3D_17927193494095
// MI455X (gfx1250) — compile-verified
//
#include <hip/hip_runtime.h>
#include <math.h>

// ---------------- static problem shape ----------------
#define BS        2
#define NQ        10000
#define NV        19560
#define EMBED     256
#define HEADS     8
#define HEAD_DIM  32
#define LEVELS    4
#define POINTS    4
#define ATTN_N    (HEADS * LEVELS * POINTS)   // 128

typedef __attribute__((ext_vector_type(2))) float v2f;
typedef __attribute__((ext_vector_type(8))) float v8f;

// ---------------- q = query + query_pos ----------------
__global__ void add_qpos_kernel(const float* __restrict__ a,
                                const float* __restrict__ b,
                                float* __restrict__ o, int n) {
    int i = blockIdx.x * blockDim.x + threadIdx.x;
    if (i < n) o[i] = a[i] + b[i];
}

// ---- fp32 WMMA GEMM: Y(M,N) = X(M,256) @ W(256,N) + bias, N compile-time --
// One wave32 per 16x64 output strip: 4 accumulators share one A tile.
// Per K-step (K=4): 1x b64 A load, 8x b32 B loads, 4x V_WMMA_F32_16X16X4_F32.
// A (16x4 f32): lane l<16 -> M=l, VGPR j -> K = 2*(l>=16)+j.
// B (4x16 f32): N = l&15,          VGPR j -> K = 2*(l>=16)+j.
// C/D: VGPR r -> M = r + 8*(l>=16), N = l&15.
template <int N>
__global__ __launch_bounds__(256) void wmma_gemm_f32_kernel(
    const float* __restrict__ X, const float* __restrict__ W,
    const float* __restrict__ bias, float* __restrict__ Y, int M) {
    const int wave   = blockIdx.x * 8 + (threadIdx.x >> 5);
    const int tilesN = N / 64;
    const int tileM  = wave / tilesN;
    const int tileN  = wave - tileM * tilesN;
    if (tileM * 16 >= M) return;              // wave-uniform: EXEC stays all-1s

    const int lane = threadIdx.x & 31;
    const int lm   = lane & 15;
    const int lh   = lane >> 4;               // K-half select

    const float* __restrict__ ap = X + (size_t)(tileM * 16 + lm) * EMBED + 2 * lh;
    const float* __restrict__ bp = W + (size_t)(2 * lh) * N + tileN * 64 + lm;

    v8f acc[4] = {{}, {}, {}, {}};
#pragma unroll 4
    for (int k = 0; k < EMBED; k += 4) {
        const v2f a = *(const v2f*)(ap + k);  // X[row, k + 2*lh + {0,1}]
        v2f b[4];
#pragma unroll
        for (int g = 0; g < 4; ++g) {         // W[k+2*lh+{0,1}, col+16*g]
            b[g].x = bp[k * N + 16 * g];
            b[g].y = bp[k * N + N + 16 * g];
        }
#pragma unroll
        for (int g = 0; g < 4; ++g)
            acc[g] = __builtin_amdgcn_wmma_f32_16x16x4_f32(
                false, a, false, b[g], (short)0, acc[g], false, false);
    }

#pragma unroll
    for (int g = 0; g < 4; ++g) {
        const int colB = tileN * 64 + 16 * g + lm;
        const float bv = bias[colB];
        float* __restrict__ yp = Y + (size_t)(tileM * 16 + lh * 8) * N + colB;
#pragma unroll
        for (int r = 0; r < 8; ++r)
            yp[(size_t)r * N] = acc[g][r] + bv;   // r*N folds to imm offsets
    }
}

// ---------------- deformable sampling ----------------
// 1 wave per workgroup (blockDim=32): one (batch, query, head) per block, so
// all per-query math derives from blockIdx -> SALU/SMEM; lanes carry the 32
// head channels; gathers are saddr-form 128B coalesced loads (L2-resident).
__global__ __launch_bounds__(32) void deform_sample_kernel(
    const float* __restrict__ vproj,   // (BS*NV, 256) = (b,pix,head,32)
    const float* __restrict__ offb,    // (BS*NQ, 256) = (h,l,p,2)
    const float* __restrict__ attnb,   // (BS*NQ, 128) = (h, l*4+p)
    const float* __restrict__ refpts,  // (BS*NQ, 4, 2)
    float* __restrict__ out) {         // (BS*NQ, 256)
    const int wave = blockIdx.x;               // 0 .. BS*NQ*HEADS-1
    const int h    = wave & (HEADS - 1);
    const int bq   = wave >> 3;
    const int lane = threadIdx.x;              // channel 0..31
    const int b    = (bq >= NQ) ? 1 : 0;       // bq / NQ (scalar select)

    const int Hs[LEVELS]  = {92, 46, 23, 12};
    const int Wd_[LEVELS] = {160, 80, 40, 20};
    const int ST[LEVELS]  = {0, 14720, 18400, 19320};

    // softmax over 16 (level,point) logits of this head (scalar loads)
    const float* ap = attnb + (size_t)bq * ATTN_N + h * 16;
    float w[16];
    float mx = -3.0e38f;
#pragma unroll
    for (int i = 0; i < 16; ++i) { w[i] = ap[i]; mx = fmaxf(mx, w[i]); }
    float s = 0.f;
#pragma unroll
    for (int i = 0; i < 16; ++i) { w[i] = __expf(w[i] - mx); s += w[i]; }
    const float inv = 1.0f / s;

    const float* op = offb + (size_t)bq * EMBED + h * (LEVELS * POINTS * 2);
    const float* rp = refpts + (size_t)bq * 8;    // (NUM_Z=4, 2); z == point

    float acc = 0.f;
#pragma unroll
    for (int l = 0; l < LEVELS; ++l) {
        const int H = Hs[l], Wd = Wd_[l];
        const float invW = 1.0f / (float)Wd, invH = 1.0f / (float)H;
        const float* lvl =
            vproj + ((size_t)(b * NV + ST[l])) * EMBED + h * HEAD_DIM;
#pragma unroll
        for (int p = 0; p < POINTS; ++p) {
            const float ox = op[(l * POINTS + p) * 2 + 0];
            const float oy = op[(l * POINTS + p) * 2 + 1];
            // loc = ref + off/(W,H); pixel = loc*dim - 0.5 (align_corners=F)
            const float x = (rp[p * 2 + 0] + ox * invW) * (float)Wd - 0.5f;
            const float y = (rp[p * 2 + 1] + oy * invH) * (float)H - 0.5f;
            const float x0f = floorf(x), y0f = floorf(y);
            const float wx1 = x - x0f, wy1 = y - y0f;
            const float wx0 = 1.f - wx1, wy0 = 1.f - wy1;
            const int x0 = (int)x0f, y0 = (int)y0f;
            const int x1 = x0 + 1, y1 = y0 + 1;
            // validity masks (zeros padding), clamped always-in-bounds indices
            const float mx0 = (x0 >= 0 && x0 < Wd) ? 1.f : 0.f;
            const float mx1 = (x1 >= 0 && x1 < Wd) ? 1.f : 0.f;
            const float my0 = (y0 >= 0 && y0 < H) ? 1.f : 0.f;
            const float my1 = (y1 >= 0 && y1 < H) ? 1.f : 0.f;
            const int cx0 = min(max(x0, 0), Wd - 1), cx1 = min(max(x1, 0), Wd - 1);
            const int cy0 = min(max(y0, 0), H - 1),  cy1 = min(max(y1, 0), H - 1);
            // force gather bases into SGPRs (values are wave-uniform)
            const int i00 = __builtin_amdgcn_readfirstlane(cy0 * Wd + cx0);
            const int i01 = __builtin_amdgcn_readfirstlane(cy0 * Wd + cx1);
            const int i10 = __builtin_amdgcn_readfirstlane(cy1 * Wd + cx0);
            const int i11 = __builtin_amdgcn_readfirstlane(cy1 * Wd + cx1);
            const float g00 = lvl[(size_t)i00 * EMBED + lane];
            const float g01 = lvl[(size_t)i01 * EMBED + lane];
            const float g10 = lvl[(size_t)i10 * EMBED + lane];
            const float g11 = lvl[(size_t)i11 * EMBED + lane];
            const float aw = w[l * POINTS + p] * inv;
            acc += aw * (wy0 * (wx0 * my0 * mx0 * g00 + wx1 * my0 * mx1 * g01) +
                         wy1 * (wx0 * my1 * mx0 * g10 + wx1 * my1 * mx1 * g11));
        }
    }
    out[(size_t)bq * EMBED + h * HEAD_DIM + lane] = acc;
}

// ---------------- launch ----------------
extern "C" void kernel_launch(void* const* d_in, const int* in_sizes, int n_in,
                              void* d_out, int out_size, void* d_ws, size_t ws_size,
                              hipStream_t stream) {
    (void)in_sizes; (void)n_in; (void)out_size; (void)ws_size;

    const float* query     = (const float*)d_in[0];
    const float* value     = (const float*)d_in[1];
    const float* query_pos = (const float*)d_in[2];
    const float* refpts    = (const float*)d_in[3];
    const float* W_val     = (const float*)d_in[4];
    const float* b_val     = (const float*)d_in[5];
    const float* W_off     = (const float*)d_in[6];
    const float* b_off     = (const float*)d_in[7];
    const float* W_attn    = (const float*)d_in[8];
    const float* b_attn    = (const float*)d_in[9];
    float* out = (float*)d_out;

    // workspace carve-up (floats): q | vproj | off | attn  (~91 MB total)
    float* q     = (float*)d_ws;
    float* vproj = q     + (size_t)BS * NQ * EMBED;
    float* offb  = vproj + (size_t)BS * NV * EMBED;
    float* attnb = offb  + (size_t)BS * NQ * EMBED;

    const int nqe = BS * NQ * EMBED;
    add_qpos_kernel<<<(nqe + 255) / 256, 256, 0, stream>>>(query, query_pos, q, nqe);

    {   // value projection: (39120 x 256) @ (256 x 256), 16x64 strips
        const int M = BS * NV;
        const int waves = (M / 16) * (EMBED / 64);           // 9780
        wmma_gemm_f32_kernel<EMBED><<<(waves + 7) / 8, 256, 0, stream>>>(
            value, W_val, b_val, vproj, M);
    }
    {   // sampling offsets: (20000 x 256) @ (256 x 256)
        const int M = BS * NQ;
        const int waves = (M / 16) * (EMBED / 64);           // 5000
        wmma_gemm_f32_kernel<EMBED><<<(waves + 7) / 8, 256, 0, stream>>>(
            q, W_off, b_off, offb, M);
    }
    {   // attention logits: (20000 x 256) @ (256 x 128)
        const int M = BS * NQ;
        const int waves = (M / 16) * (ATTN_N / 64);          // 2500
        wmma_gemm_f32_kernel<ATTN_N><<<(waves + 7) / 8, 256, 0, stream>>>(
            q, W_attn, b_attn, attnb, M);
    }

    // one wave32 workgroup per (b, query, head)
    deform_sample_kernel<<<BS * NQ * HEADS, 32, 0, stream>>>(
        vproj, offb, attnb, refpts, out);
}